// SheafLaplacian_71743133713063
// MI455X (gfx1250) — compile-verified
//
#include <hip/hip_runtime.h>

typedef __attribute__((ext_vector_type(2))) float v2f;
typedef __attribute__((ext_vector_type(8))) float v8f;

#define SHEAF_ALPHA 0.1f

// ---------------------------------------------------------------------------
// out = x  (identity part of (I - alpha*L) x); the edge kernel accumulates
// -alpha * Lx with hardware f32 atomics. out (25.6MB) stays L2-resident.
// ---------------------------------------------------------------------------
__global__ __launch_bounds__(256) void sheaf_init_out(const float4* __restrict__ x4,
                                                      float4* __restrict__ out4,
                                                      int n4) {
  int i = blockIdx.x * blockDim.x + threadIdx.x;
  int stride = gridDim.x * blockDim.x;
  for (; i < n4; i += stride) out4[i] = x4[i];
}

// ---------------------------------------------------------------------------
// One edge per wave32.
//   Stage 1: msg(8x8)      = [F1 | -F0](8x16) @ [Xv ; Xu](16x8)   (4 WMMA K4)
//   Stage 2: contrib(16x8) = [F1^T ; -F0^T](16x8) @ msg(8x8)      (2 WMMA K4)
// All loads are unconditional with clamped (duplicated) indices: the
// duplicate products land in msg rows 8-15 / cols 8-15 and C2 cols 8-15,
// which are never consumed. No EXEC divergence until the atomic scatter.
//
// A layout (32-bit 16x4): lanes 0-15 hold M=0..15, K={0,1}; lanes 16-31 hold
// K={2,3}. B is the transposed analogue. C/D: lane n = column n, VGPR r =
// row r (lanes 16-31: rows 8-15).
// ---------------------------------------------------------------------------
__global__ __launch_bounds__(256) void sheaf_edge_kernel(
    const float* __restrict__ x,      // [8, N_V, 8]
    const float* __restrict__ rmaps,  // [N_E, 2, 8, 8]  ([:,0]=F0, [:,1]=F1)
    const int*   __restrict__ edges,  // [N_E, 2]
    float*       __restrict__ out,    // [8, N_V, 8]
    int n_e, int n_v)
{
  const int wave = (int)((blockIdx.x * (unsigned)blockDim.x + threadIdx.x) >> 5);
  const int lane = (int)(threadIdx.x & 31u);
  if (wave >= n_e) return;             // wave-granular: EXEC stays all-1s

  const int e = wave;
  const int u = edges[2 * e + 0];
  const int v = edges[2 * e + 1];

  const int row = lane & 15;           // A row / B batch column (clamped below)
  const int kh  = lane >> 4;           // K-half of each K=4 WMMA step
  const int b   = row & 7;             // clamped batch / matrix row
  const int c0  = 2 * kh;              // first K element this lane supplies

  const unsigned fbase = (unsigned)e * 128u;                 // floats
  const float* __restrict__ F0 = rmaps + fbase;              // 8x8 row-major
  const float* __restrict__ F1 = rmaps + (fbase + 64u);

  // ---- stage-1 A fragments: rows of [F1 | -F0] (v2f, 8B aligned, NT) ----
  const unsigned r8 = (unsigned)b * 8u + (unsigned)c0;
  v2f a0 = __builtin_nontemporal_load((const v2f*)(F1 + r8));        // K 0..3
  v2f a1 = __builtin_nontemporal_load((const v2f*)(F1 + r8 + 4));    // K 4..7
  v2f a2 = __builtin_nontemporal_load((const v2f*)(F0 + r8));        // K 8..11
  v2f a3 = __builtin_nontemporal_load((const v2f*)(F0 + r8 + 4));    // K 12..15
  a2 = -a2;                                                          // -F0
  a3 = -a3;

  // ---- stage-2 A fragments: columns of [F1^T ; -F0^T] (scalar, NT) ----
  const float* __restrict__ G  = rmaps + (fbase + ((row < 8) ? 64u : 0u));
  const float  gs = (row < 8) ? 1.0f : -1.0f;
  float g00 = __builtin_nontemporal_load(&G[(c0    ) * 8 + b]);      // K=c0
  float g01 = __builtin_nontemporal_load(&G[(c0 + 1) * 8 + b]);      // K=c0+1
  float g10 = __builtin_nontemporal_load(&G[(c0 + 4) * 8 + b]);      // K=c0+4
  float g11 = __builtin_nontemporal_load(&G[(c0 + 5) * 8 + b]);      // K=c0+5

  // ---- stage-1 B fragments: [Xv ; Xu] (32-bit offsets -> saddr form) ----
  const unsigned bn = (unsigned)b * (unsigned)n_v;
  const unsigned ov = (bn + (unsigned)v) * 8u + (unsigned)c0;
  const unsigned ou = (bn + (unsigned)u) * 8u + (unsigned)c0;
  v2f b0 = *(const v2f*)(x + ov);       // Xv, K 0..3
  v2f b1 = *(const v2f*)(x + ov + 4);   // Xv, K 4..7
  v2f b2 = *(const v2f*)(x + ou);       // Xu, K 8..11
  v2f b3 = *(const v2f*)(x + ou + 4);   // Xu, K 12..15

  // ---- Stage 1: msg = [F1 | -F0] @ [Xv ; Xu]  (4 back-to-back WMMAs) ----
  v8f msg = {};
  msg = __builtin_amdgcn_wmma_f32_16x16x4_f32(false, a0, false, b0, (short)0, msg, false, false);
  msg = __builtin_amdgcn_wmma_f32_16x16x4_f32(false, a1, false, b1, (short)0, msg, false, false);
  msg = __builtin_amdgcn_wmma_f32_16x16x4_f32(false, a2, false, b2, (short)0, msg, false, false);
  msg = __builtin_amdgcn_wmma_f32_16x16x4_f32(false, a3, false, b3, (short)0, msg, false, false);

  // ---- Stage 2: contrib = [F1^T ; -F0^T] @ msg ----
  // Re-layout msg from C layout to B layout with wave32 shuffles.
  const int src = row;
  v8f ctr = {};
  {
    const float t0 = __shfl(msg[0], src, 32);
    const float t1 = __shfl(msg[1], src, 32);
    const float t2 = __shfl(msg[2], src, 32);
    const float t3 = __shfl(msg[3], src, 32);
    v2f bb; bb.x = kh ? t2 : t0; bb.y = kh ? t3 : t1;
    v2f aa; aa.x = gs * g00;     aa.y = gs * g01;
    ctr = __builtin_amdgcn_wmma_f32_16x16x4_f32(false, aa, false, bb, (short)0, ctr, false, false);
  }
  {
    const float t0 = __shfl(msg[4], src, 32);
    const float t1 = __shfl(msg[5], src, 32);
    const float t2 = __shfl(msg[6], src, 32);
    const float t3 = __shfl(msg[7], src, 32);
    v2f bb; bb.x = kh ? t2 : t0; bb.y = kh ? t3 : t1;
    v2f aa; aa.x = gs * g10;     aa.y = gs * g11;
    ctr = __builtin_amdgcn_wmma_f32_16x16x4_f32(false, aa, false, bb, (short)0, ctr, false, false);
  }

  // ---- Scatter: out[b, node, j] += -alpha * contrib (hardware f32 atomics)
  // lanes 0-7: contrib_v for batches 0-7; lanes 16-23: contrib_u.
  if (row < 8) {
    const int node = (lane < 16) ? v : u;
    float* orow = out + (size_t)((bn + (unsigned)node) * 8u);
#pragma unroll
    for (int j = 0; j < 8; ++j)
      unsafeAtomicAdd(&orow[j], -SHEAF_ALPHA * ctr[j]);
  }
}

extern "C" void kernel_launch(void* const* d_in, const int* in_sizes, int n_in,
                              void* d_out, int out_size, void* d_ws, size_t ws_size,
                              hipStream_t stream) {
  const float* x     = (const float*)d_in[0];
  const float* rmaps = (const float*)d_in[1];
  const int*   edges = (const int*)d_in[2];
  float*       out   = (float*)d_out;

  const int n_x = in_sizes[0];       // 8 * N_V * 8
  const int n_e = in_sizes[2] / 2;   // edges is [N_E, 2]
  const int n_v = n_x / 64;          // BATCH=8, D_V=8

  // out = x
  const int n4 = n_x / 4;
  int cblocks = (n4 + 255) / 256;
  if (cblocks > 8192) cblocks = 8192;  // grid-stride copy
  sheaf_init_out<<<cblocks, 256, 0, stream>>>((const float4*)x, (float4*)out, n4);

  // one edge per wave32, 8 waves per 256-thread block
  const long long threads = (long long)n_e * 32;
  const int eblocks = (int)((threads + 255) / 256);
  sheaf_edge_kernel<<<eblocks, 256, 0, stream>>>(x, rmaps, edges, out, n_e, n_v);
}